// Qwen3OmniMoeTalkerCodePredictorAttention_28638841930217
// MI455X (gfx1250) — compile-verified
//
#include <hip/hip_runtime.h>
#include <math.h>

// ---- problem constants (match reference) ----
#define B_    16
#define S_    16
#define H_    2048
#define NH_   16
#define NKV_  4
#define G_    4
#define D_    128
#define MAXC_ 4352
#define NCHUNK 4
#define EPS_  1e-6f
#define TT    32          // K/V tile rows staged in LDS per iteration

#if __has_builtin(__builtin_amdgcn_global_load_async_to_lds_b128) && \
    __has_builtin(__builtin_amdgcn_s_wait_asynccnt)
#define HAVE_ASYNC 1
#else
#define HAVE_ASYNC 0
#endif

typedef float v2f __attribute__((ext_vector_type(2)));
typedef float v4f __attribute__((ext_vector_type(4)));
typedef float v8f __attribute__((ext_vector_type(8)));
typedef int   i128 __attribute__((vector_size(16)));  // 16B payload type

__device__ __forceinline__ v8f vzero8() {
  v8f r;
#pragma unroll
  for (int i = 0; i < 8; ++i) r[i] = 0.0f;
  return r;
}

// D = A(16x4) * B(4x16) + C, f32 WMMA.
// A frag: lanes 0-15 hold M=lane, K=0..1; lanes 16-31 K=2..3 (v2f)
// B frag: v0 = rows K=0 / K=2 (half-wave), v1 = K=1 / K=3, N = lane%16
// C/D:    VGPR j: M = j (+8 for upper half-wave), N = lane%16
__device__ __forceinline__ v8f wmma_f32(v2f a, v2f b, v8f c) {
  return __builtin_amdgcn_wmma_f32_16x16x4_f32(false, a, false, b,
                                               (short)0, c, false, false);
}

#if HAVE_ASYNC
// 16B global -> LDS direct (no VGPR round trip), tracked by ASYNCcnt
__device__ __forceinline__ void async_cp16(const float* g, float* l) {
  __builtin_amdgcn_global_load_async_to_lds_b128(
      (__attribute__((address_space(1))) i128*)g,
      (__attribute__((address_space(3))) i128*)l, 0, 0);
}
#endif

// -------------------------------------------------------------------------
// C(MxN) = X(MxK) @ W^T, W is (N x K) row-major. One wave per 16-col strip,
// MT row-tiles per wave (B-fragment reuse). f32 WMMA 16x16x4.
// -------------------------------------------------------------------------
template <int MT>
__global__ __launch_bounds__(32) void gemm_xwt(const float* __restrict__ X,
                                               const float* __restrict__ W,
                                               float* __restrict__ C,
                                               int M, int N, int K) {
  const int lane = threadIdx.x & 31;
  const int hlf  = lane >> 4;
  const int l16  = lane & 15;
  const int n0   = blockIdx.x * 16;
  const int m0   = blockIdx.y * (16 * MT);

  const float* wrow = W + (size_t)(n0 + l16) * K + 2 * hlf;
  const float* xrow[MT];
#pragma unroll
  for (int t = 0; t < MT; ++t)
    xrow[t] = X + (size_t)(m0 + t * 16 + l16) * K + 2 * hlf;

  v8f acc[MT];
#pragma unroll
  for (int t = 0; t < MT; ++t) acc[t] = vzero8();

  for (int k = 0; k < K; k += 4) {
    if ((k & 127) == 0)  // stream weights ahead (global_prefetch_b8)
      __builtin_prefetch(wrow + k + 128, 0, 1);
    v2f b = *(const v2f*)(wrow + k);
#pragma unroll
    for (int t = 0; t < MT; ++t) {
      v2f a  = *(const v2f*)(xrow[t] + k);
      acc[t] = wmma_f32(a, b, acc[t]);
    }
  }

#pragma unroll
  for (int t = 0; t < MT; ++t)
#pragma unroll
    for (int j = 0; j < 8; ++j)
      C[(size_t)(m0 + t * 16 + j + 8 * hlf) * N + n0 + l16] = acc[t][j];
}

// -------------------------------------------------------------------------
// Fused RMSNorm + RoPE + head-major transpose. One wave per 128-d vector.
//  vid <  4096           : Q rows  -> Qr (B,NKV,G,S,D)
//  4096 <= vid < 5120    : K rows  -> Kr (B,NKV,S,D)
//  5120 <= vid < 6144    : V rows  -> Vr (B,NKV,S,D)   (copy only)
// -------------------------------------------------------------------------
__global__ __launch_bounds__(128) void normrope(
    const float* __restrict__ Qraw, const float* __restrict__ Kraw,
    const float* __restrict__ Vraw, const float* __restrict__ cosd,
    const float* __restrict__ sind, const float* __restrict__ qw,
    const float* __restrict__ kw, float* __restrict__ Qr,
    float* __restrict__ Kr, float* __restrict__ Vr) {
  const int wave = threadIdx.x >> 5;
  const int lane = threadIdx.x & 31;
  const int hlf  = lane >> 4;
  const int vid  = blockIdx.x * 4 + wave;  // 0..6143 (uniform per wave)
  const int d0   = lane * 4;

  const float* src;
  const float* nw = nullptr;
  float* dst;
  int r;
  bool dorope;

  if (vid < 4096) {  // Q
    r            = vid >> 4;
    const int hh = vid & 15;
    src = Qraw + (size_t)r * (NH_ * D_) + hh * D_;
    const int b = r >> 4, s = r & 15;
    dst = Qr + ((((size_t)b * NKV_ + (hh >> 2)) * G_ + (hh & 3)) * S_ + s) * D_;
    nw = qw;  dorope = true;
  } else if (vid < 5120) {  // K
    const int ki = vid - 4096;
    r            = ki >> 2;
    const int hh = ki & 3;
    src = Kraw + (size_t)r * (NKV_ * D_) + hh * D_;
    const int b = r >> 4, s = r & 15;
    dst = Kr + (((size_t)b * NKV_ + hh) * S_ + s) * D_;
    nw = kw;  dorope = true;
  } else {  // V (plain transpose copy)
    const int vi = vid - 5120;
    r            = vi >> 2;
    const int hh = vi & 3;
    src = Vraw + (size_t)r * (NKV_ * D_) + hh * D_;
    const int b = r >> 4, s = r & 15;
    dst = Vr + (((size_t)b * NKV_ + hh) * S_ + s) * D_;
    dorope = false;
  }

  v4f xv = *(const v4f*)(src + d0);
  if (!dorope) {
    *(v4f*)(dst + d0) = xv;
    return;
  }

  float ss = xv.x * xv.x + xv.y * xv.y + xv.z * xv.z + xv.w * xv.w;
#pragma unroll
  for (int o = 16; o >= 1; o >>= 1) ss += __shfl_xor(ss, o, 32);
  const float rms = rsqrtf(ss * (1.0f / 128.0f) + EPS_);

  const float* cp = cosd + (size_t)r * D_ + d0;
  const float* sp = sind + (size_t)r * D_ + d0;
  float out[4];
#pragma unroll
  for (int i = 0; i < 4; ++i) {
    const float xn   = xv[i] * rms * nw[d0 + i];
    const float part = __shfl_xor(xn, 16, 32);       // value at d ^ 64
    const float rot  = (hlf == 0) ? -part : part;    // rotate_half
    out[i] = xn * cp[i] + rot * sp[i];
  }
  *(v4f*)(dst + d0) = (v4f){out[0], out[1], out[2], out[3]};
}

// -------------------------------------------------------------------------
// Flash-decoding partial attention. Grid = B*NKV*NCHUNK workgroups of 128
// threads (4 waves). Wave w owns GQA group g=w (16 queries); all 4 waves
// share each K/V tile staged in LDS (4x reuse -> each cached byte is read
// from HBM exactly once across the whole dispatch). K/V tiles are double
// buffered and streamed with GLOBAL_LOAD_ASYNC_TO_LDS_B128 (ASYNCcnt) so
// tile c+1 is in flight behind the WMMA work on tile c. Q lives in
// registers as prebuilt WMMA B-fragments. Online softmax on transposed
// score tiles.
// -------------------------------------------------------------------------
__global__ __launch_bounds__(128) void attn_partial(
    const float* __restrict__ cached_k, const float* __restrict__ cached_v,
    const float* __restrict__ Qr, const float* __restrict__ Kr,
    const float* __restrict__ Vr, const int* __restrict__ cache_len_p,
    float* __restrict__ Opart, float* __restrict__ Mpart,
    float* __restrict__ Lpart) {
  __shared__ float Ks[2][TT * 132];   // padded: stride 132 spreads banks
  __shared__ float Vs[2][TT * 132];
  __shared__ float Pb[4 * 16 * 18];   // per-wave P tile (q x t), stride 18

  const int tid  = threadIdx.x;
  const int wave = tid >> 5;          // == GQA group g
  const int lane = tid & 31;
  const int hlf  = lane >> 4;
  const int l16  = lane & 15;

  const int bid       = blockIdx.x;              // (b*NKV+kv)*NCHUNK + chunk
  const int chunk     = bid & (NCHUNK - 1);
  const int kvid      = bid / NCHUNK;            // b*NKV + kv
  const int cache_len = *cache_len_p;
  const int chunk_len = cache_len / NCHUNK;      // 1024
  const int ntiles    = chunk_len / TT;          // 32
  const int t_begin   = chunk * chunk_len;

  // Q^T fragments for this wave's 16 queries, kept in VGPRs for the whole T loop
  const float* qp = Qr + ((size_t)kvid * G_ + wave) * (S_ * D_) + l16 * D_ + 2 * hlf;
  v2f bq[32];
#pragma unroll
  for (int f = 0; f < 32; ++f) bq[f] = *(const v2f*)(qp + 4 * f);

  const float* Kb = cached_k + (size_t)kvid * MAXC_ * D_;
  const float* Vb = cached_v + (size_t)kvid * MAXC_ * D_;

  float m_run = -__builtin_inff();
  float l_run = 0.0f;
  v8f acc[8];
#pragma unroll
  for (int dt = 0; dt < 8; ++dt) acc[dt] = vzero8();

  const float scaling = 0.0883883476483184f;  // 1/sqrt(128)
  const int niter = ntiles + ((chunk == NCHUNK - 1) ? 1 : 0);  // +1: new tokens

  // stage tile c into buffer c&1 (async direct-to-LDS when available)
  auto stage_tile = [&](int c) {
    float* kdst = Ks[c & 1];
    float* vdst = Vs[c & 1];
    if (c < ntiles) {
      const int tbase = t_begin + c * TT;
      for (int i = tid; i < TT * 32; i += 128) {
        const int row = i >> 5, seg = i & 31;
        const float* kg = Kb + (size_t)(tbase + row) * D_ + seg * 4;
        const float* vg = Vb + (size_t)(tbase + row) * D_ + seg * 4;
#if HAVE_ASYNC
        async_cp16(kg, &kdst[row * 132 + seg * 4]);
        async_cp16(vg, &vdst[row * 132 + seg * 4]);
#else
        *(v4f*)&kdst[row * 132 + seg * 4] = *(const v4f*)kg;
        *(v4f*)&vdst[row * 132 + seg * 4] = *(const v4f*)vg;
#endif
      }
    } else {  // epilogue: the S_ freshly projected tokens
      for (int i = tid; i < S_ * 32; i += 128) {
        const int row = i >> 5, seg = i & 31;
        const float* kg = Kr + ((size_t)kvid * S_ + row) * D_ + seg * 4;
        const float* vg = Vr + ((size_t)kvid * S_ + row) * D_ + seg * 4;
#if HAVE_ASYNC
        async_cp16(kg, &kdst[row * 132 + seg * 4]);
        async_cp16(vg, &vdst[row * 132 + seg * 4]);
#else
        *(v4f*)&kdst[row * 132 + seg * 4] = *(const v4f*)kg;
        *(v4f*)&vdst[row * 132 + seg * 4] = *(const v4f*)vg;
#endif
      }
    }
  };

  stage_tile(0);

  for (int c = 0; c < niter; ++c) {
#if HAVE_ASYNC
    __builtin_amdgcn_s_wait_asynccnt(0);  // my in-flight tile is resident
#endif
    __syncthreads();                      // everyone's tile is resident
    if (c + 1 < niter) stage_tile(c + 1); // prefetch next buffer behind compute

    const int cur  = c & 1;
    const int rows = (c < ntiles) ? TT : S_;
    const float* ks = Ks[cur];
    const float* vs = Vs[cur];

    for (int kt0 = 0; kt0 < rows; kt0 += 16) {
      // ---- S^T tile (16 t-rows x 16 queries) = K_tile(16x128) x Q^T ----
      v8f st = vzero8();
#pragma unroll
      for (int f = 0; f < 32; ++f) {
        v2f a = *(const v2f*)&ks[(kt0 + l16) * 132 + 4 * f + 2 * hlf];
        st = wmma_f32(a, bq[f], st);
      }
      // lane holds q=l16, t_local = j + 8*hlf across the 8 acc VGPRs
      float mc = -__builtin_inff();
#pragma unroll
      for (int j = 0; j < 8; ++j) {
        st[j] *= scaling;
        mc = fmaxf(mc, st[j]);
      }
      mc = fmaxf(mc, __shfl_xor(mc, 16, 32));  // combine t-halves per query
      const float m_new = fmaxf(m_run, mc);
      const float alpha = __expf(m_run - m_new);
      float ls = 0.0f;
#pragma unroll
      for (int j = 0; j < 8; ++j) {
        st[j] = __expf(st[j] - m_new);
        ls += st[j];
      }
      ls += __shfl_xor(ls, 16, 32);
      l_run = l_run * alpha + ls;
      m_run = m_new;

      // rescale accumulators (alpha indexed by q = j + 8*hlf) and park P in LDS
#pragma unroll
      for (int j = 0; j < 8; ++j) {
        const float aq = __shfl(alpha, j + 8 * hlf, 32);
#pragma unroll
        for (int dt = 0; dt < 8; ++dt) acc[dt][j] *= aq;
        Pb[(wave * 16 + l16) * 18 + j + 8 * hlf] = st[j];  // P[q][t]
      }

      // ---- acc(16q x 128d) += P(16x16) x V_tile(16x128) ----
#pragma unroll
      for (int dt = 0; dt < 8; ++dt) {
#pragma unroll
        for (int k0 = 0; k0 < 16; k0 += 4) {
          v2f a = *(const v2f*)&Pb[(wave * 16 + l16) * 18 + k0 + 2 * hlf];
          v2f bv;
          bv.x = vs[(kt0 + k0 + 2 * hlf) * 132 + dt * 16 + l16];
          bv.y = vs[(kt0 + k0 + 2 * hlf + 1) * 132 + dt * 16 + l16];
          acc[dt] = wmma_f32(a, bv, acc[dt]);
        }
      }
    }
  }

  // ---- emit per-chunk partials: running max, sum, unnormalized output ----
  const size_t pbase = ((size_t)kvid * NCHUNK + chunk) * (G_ * S_);
  if (lane < 16) {
    Mpart[pbase + wave * S_ + l16] = m_run;
    Lpart[pbase + wave * S_ + l16] = l_run;
  }
  float* op = Opart + (pbase + wave * S_) * D_;
#pragma unroll
  for (int dt = 0; dt < 8; ++dt)
#pragma unroll
    for (int j = 0; j < 8; ++j)
      op[(size_t)(j + 8 * hlf) * D_ + dt * 16 + l16] = acc[dt][j];
}

// -------------------------------------------------------------------------
// Merge the NCHUNK flash partials. One block per (b,kv,g,s), thread = d.
// -------------------------------------------------------------------------
__global__ __launch_bounds__(128) void attn_combine(
    const float* __restrict__ Opart, const float* __restrict__ Mpart,
    const float* __restrict__ Lpart, float* __restrict__ attn) {
  const int cid  = blockIdx.x;  // ((b*NKV+kv)*G + g)*S + s
  const int d    = threadIdx.x;
  const int s    = cid & 15;
  const int g    = (cid >> 4) & 3;
  const int kvid = cid >> 6;
  const int b    = kvid >> 2;
  const int kv   = kvid & 3;

  float mrow[NCHUNK], lrow[NCHUNK], e[NCHUNK];
  float M = -__builtin_inff();
#pragma unroll
  for (int j = 0; j < NCHUNK; ++j) {
    const size_t idx = ((size_t)kvid * NCHUNK + j) * (G_ * S_) + g * S_ + s;
    mrow[j] = Mpart[idx];
    lrow[j] = Lpart[idx];
    M = fmaxf(M, mrow[j]);
  }
  float L = 0.0f;
#pragma unroll
  for (int j = 0; j < NCHUNK; ++j) {
    e[j] = __expf(mrow[j] - M);
    L += lrow[j] * e[j];
  }
  const float invL = 1.0f / L;
  float o = 0.0f;
#pragma unroll
  for (int j = 0; j < NCHUNK; ++j) {
    const size_t idx = ((size_t)kvid * NCHUNK + j) * (G_ * S_) + g * S_ + s;
    o += Opart[idx * D_ + d] * e[j];
  }
  const int h = kv * G_ + g;
  attn[(((size_t)b * S_ + s) * NH_ + h) * D_ + d] = o * invL;
}

// -------------------------------------------------------------------------
extern "C" void kernel_launch(void* const* d_in, const int* in_sizes, int n_in,
                              void* d_out, int out_size, void* d_ws,
                              size_t ws_size, hipStream_t stream) {
  (void)in_sizes; (void)n_in; (void)out_size; (void)ws_size;
  const float* hs   = (const float*)d_in[0];
  const float* cosd = (const float*)d_in[1];
  const float* sind = (const float*)d_in[2];
  const float* ck   = (const float*)d_in[3];
  const float* cv   = (const float*)d_in[4];
  const float* wq   = (const float*)d_in[5];
  const float* wk   = (const float*)d_in[6];
  const float* wv   = (const float*)d_in[7];
  const float* wo   = (const float*)d_in[8];
  const float* qw   = (const float*)d_in[9];
  const float* kw   = (const float*)d_in[10];
  const int*   clen = (const int*)d_in[11];

  float* ws    = (float*)d_ws;
  float* Qraw  = ws; ws += 256 * 2048;
  float* Kraw  = ws; ws += 256 * 512;
  float* Vraw  = ws; ws += 256 * 512;
  float* Qr    = ws; ws += 256 * 2048;
  float* Kr    = ws; ws += 256 * 512;
  float* Vr    = ws; ws += 256 * 512;
  float* Opart = ws; ws += (size_t)B_ * NKV_ * NCHUNK * G_ * S_ * D_;
  float* Mpart = ws; ws += (size_t)B_ * NKV_ * NCHUNK * G_ * S_;
  float* Lpart = ws; ws += (size_t)B_ * NKV_ * NCHUNK * G_ * S_;
  float* attnb = ws; ws += 256 * 2048;

  // QKV projections (f32 WMMA GEMMs)
  gemm_xwt<4><<<dim3(128, 4), 32, 0, stream>>>(hs, wq, Qraw, 256, 2048, 2048);
  gemm_xwt<4><<<dim3(32, 4), 32, 0, stream>>>(hs, wk, Kraw, 256, 512, 2048);
  gemm_xwt<4><<<dim3(32, 4), 32, 0, stream>>>(hs, wv, Vraw, 256, 512, 2048);

  // RMSNorm + RoPE + transpose to head-major
  normrope<<<1536, 128, 0, stream>>>(Qraw, Kraw, Vraw, cosd, sind, qw, kw,
                                     Qr, Kr, Vr);

  // chunked flash attention + combine
  attn_partial<<<B_ * NKV_ * NCHUNK, 128, 0, stream>>>(ck, cv, Qr, Kr, Vr,
                                                       clen, Opart, Mpart,
                                                       Lpart);
  attn_combine<<<B_ * NKV_ * G_ * S_, 128, 0, stream>>>(Opart, Mpart, Lpart,
                                                        attnb);

  // output projection straight into d_out
  gemm_xwt<4><<<dim3(128, 4), 32, 0, stream>>>(attnb, wo, (float*)d_out,
                                               256, 2048, 2048);
}